// SplineInterpolation_86569360818325
// MI455X (gfx1250) — compile-verified
//
#include <hip/hip_runtime.h>
#include <math.h>

// ---------------------------------------------------------------------------
// Thin-plate-spline interpolation, MI455X (gfx1250, wave32).
//
// Stage 1 (tps_solve_kernel, 1 block): build + solve the 67x67 TPS system
// with partial-pivoting Gaussian elimination (the K diagonal and 3x3 corner
// are zero -> pivoting mandatory). wv [67x2] -> workspace.
//
// Stage 2 (tps_eval_kernel): each wave evaluates 16 consecutive pixels.
//     out[16x2] = A[16x68] @ B[68x16],  A = [r_0..r_63, 1, x, y, 0-pad]
//     B cols 0,1 = wv, cols 2..15 = 0 (zero-padded table in LDS)
// r_i via raw v_sqrt_f32 (__builtin_amdgcn_sqrtf, ~1 ULP — the TRANS pipe is
// the roofline ceiling: 64 sqrts/pixel * 1M pixels). The 68-deep x2
// accumulation runs on the matrix pipe as 17 chained V_WMMA_F32_16X16X4_F32
// (full f32), co-executing with the VALU/TRANS distance math.
//
// WMMA operand layouts (ISA 7.12.2, 32-bit, wave32):
//   A 16x4 : v0 = K0 (lanes 0-15) / K2 (lanes 16-31); v1 = K1 / K3
//   B 4x16 : v0 = K0 row (lanes 0-15 = N0..15) / K2 row; v1 = K1 / K3
//   D 16x16: lane L, VGPR v -> D[v + (L>=16 ? 8 : 0)][L%16]
// ---------------------------------------------------------------------------

#define OUT_W 1024
#define OUT_H 1024
#define NP 64          // control points
#define NA 67          // NP + 3 (affine)
#define NC 69          // augmented columns: 67 matrix + 2 RHS
#define NK 68          // padded K depth (17 chunks of 4)

typedef __attribute__((ext_vector_type(2))) float v2f;
typedef __attribute__((ext_vector_type(8))) float v8f;

__device__ __forceinline__ float fsqrt(float x) {
    return __builtin_amdgcn_sqrtf(x);   // raw v_sqrt_f32, no IEEE fixup chain
}

// ------------------------------ Stage 1 ------------------------------------
__global__ void tps_solve_kernel(const float* __restrict__ pts,
                                 const float* __restrict__ vals,
                                 float* __restrict__ wv) {
    __shared__ float A[NA][NC];
    __shared__ int s_piv;
    const int tid = threadIdx.x;

    // Build augmented system [X | Y] in LDS.
    for (int idx = tid; idx < NA * NC; idx += blockDim.x) {
        int r = idx / NC, c = idx % NC;
        float v = 0.0f;
        if (r < NP) {
            if (c < NP) {
                float dx = pts[2 * r] - pts[2 * c];
                float dy = pts[2 * r + 1] - pts[2 * c + 1];
                v = fsqrt(dx * dx + dy * dy);
            } else if (c == NP)     v = 1.0f;
            else if (c == NP + 1)   v = pts[2 * r];
            else if (c == NP + 2)   v = pts[2 * r + 1];
            else                    v = vals[2 * r + (c - NA)];   // RHS cols
        } else {
            if (c < NP) {
                if (r == NP)          v = 1.0f;
                else if (r == NP + 1) v = pts[2 * c];
                else                  v = pts[2 * c + 1];
            }   // else zero (3x3 corner, zero RHS tail)
        }
        A[r][c] = v;
    }
    __syncthreads();

    // Forward elimination with partial pivoting.
    for (int p = 0; p < NA; ++p) {
        if (tid == 0) {
            int best = p;
            float bv = fabsf(A[p][p]);
            for (int r = p + 1; r < NA; ++r) {
                float av = fabsf(A[r][p]);
                if (av > bv) { bv = av; best = r; }
            }
            s_piv = best;
        }
        __syncthreads();
        const int piv = s_piv;
        if (piv != p && tid < NC) {
            float t = A[p][tid]; A[p][tid] = A[piv][tid]; A[piv][tid] = t;
        }
        __syncthreads();
        const int r = p + 1 + tid;          // one thread per row below pivot
        if (r < NA) {
            float f = A[r][p] / A[p][p];
            for (int j = p; j < NC; ++j) A[r][j] -= f * A[p][j];
        }
        __syncthreads();
    }

    // Back substitution: one thread per RHS column.
    if (tid < 2) {
        const int c = NA + tid;
        for (int p = NA - 1; p >= 0; --p) {
            float v = A[p][c];
            for (int j = p + 1; j < NA; ++j) v -= A[p][j] * A[j][c];
            v /= A[p][p];
            A[p][c] = v;
            wv[2 * p + tid] = v;            // wv row-major [67][2]
        }
    }
}

// ------------------------------ Stage 2 ------------------------------------
// 256 threads = 8 waves per block; wave handles 16 consecutive pixels;
// block handles 128 consecutive pixels (one image row spans 8 blocks).
__global__ void __launch_bounds__(256)
tps_eval_kernel(const float* __restrict__ pts,
                const float* __restrict__ wv,
                float* __restrict__ out) {
    __shared__ float s_pts[2 * NP];       // [64][2] control points
    __shared__ float s_B[NK * 16];        // zero-padded B: [68 K-rows][16 N]
    const int tid = threadIdx.x;

    if (tid < 2 * NP) s_pts[tid] = pts[tid];
    // Padded B table: col 0,1 = wv (rows 0..66), everything else zero.
    // Unconditional in-loop ds_loads, no EXEC divergence, no cndmask.
    for (int idx = tid; idx < NK * 16; idx += 256) {
        int k = idx >> 4, n = idx & 15;
        s_B[idx] = (n < 2 && k < NA) ? wv[2 * k + n] : 0.0f;
    }
    __syncthreads();

    const int lane = tid & 31;
    const int wave = tid >> 5;
    const int px0  = blockIdx.x * 128 + wave * 16;  // first pixel of the tile
    const int m    = lane & 15;                     // A-row / B-col index
    const bool hi  = lane >= 16;                    // upper half-wave: K+2,K+3

    const float myX = (float)((px0 + m) & (OUT_W - 1));
    const float myY = (float)(px0 >> 10);           // whole tile on one row
    const int koff  = hi ? 2 : 0;

    v8f acc = {};
    // 16 RBF K-chunks of 4.  Per chunk this lane covers K rows k0, k0+1.
    #pragma unroll 4
    for (int c = 0; c < 16; ++c) {
        const int k0 = c * 4 + koff;
        const float px0f = s_pts[2 * k0 + 0];
        const float py0f = s_pts[2 * k0 + 1];
        const float px1f = s_pts[2 * k0 + 2];
        const float py1f = s_pts[2 * k0 + 3];
        const float dx0 = myX - px0f, dy0 = myY - py0f;
        const float dx1 = myX - px1f, dy1 = myY - py1f;
        v2f Aop = { fsqrt(dx0 * dx0 + dy0 * dy0),   // raw v_sqrt_f32 (TRANS)
                    fsqrt(dx1 * dx1 + dy1 * dy1) };
        v2f Bop = { s_B[k0 * 16 + m],
                    s_B[(k0 + 1) * 16 + m] };
        acc = __builtin_amdgcn_wmma_f32_16x16x4_f32(
            false, Aop, false, Bop, (short)0, acc, false, false);
    }

    // Affine tail chunk: K rows 64 -> 1, 65 -> x, 66 -> y, 67 -> 0.
    {
        const int k0 = 64 + koff;
        v2f Aop = { hi ? myY  : 1.0f,
                    hi ? 0.0f : myX };
        v2f Bop = { s_B[k0 * 16 + m],
                    s_B[(k0 + 1) * 16 + m] };   // row 67 is zero in the table
        acc = __builtin_amdgcn_wmma_f32_16x16x4_f32(
            false, Aop, false, Bop, (short)0, acc, false, false);
    }

    // D layout: lane L, VGPR v holds D[v + (L>=16 ? 8 : 0)][L%16].
    // Useful results live in lanes 0,1 (pixels 0-7) and 16,17 (pixels 8-15).
    if (m < 2) {
        const int mbase = hi ? 8 : 0;
        const int ch = m;                           // 0 = x-out, 1 = y-out
#pragma unroll
        for (int v = 0; v < 8; ++v) {
            const int pix = px0 + mbase + v;
            out[(size_t)pix * 2 + ch] = acc[v];
        }
    }
}

// ------------------------------ Launch -------------------------------------
extern "C" void kernel_launch(void* const* d_in, const int* in_sizes, int n_in,
                              void* d_out, int out_size, void* d_ws, size_t ws_size,
                              hipStream_t stream) {
    const float* pts  = (const float*)d_in[0];   // [64,2] f32
    const float* vals = (const float*)d_in[1];   // [64,2] f32
    float* out = (float*)d_out;                  // [1024,1024,2] f32
    float* wv  = (float*)d_ws;                   // [67,2] f32 scratch

    hipLaunchKernelGGL(tps_solve_kernel, dim3(1), dim3(256), 0, stream,
                       pts, vals, wv);

    const int pixels = OUT_W * OUT_H;
    hipLaunchKernelGGL(tps_eval_kernel, dim3(pixels / 128), dim3(256), 0, stream,
                       pts, wv, out);
}